// BMamba_18880676233418
// MI455X (gfx1250) — compile-verified
//
#include <hip/hip_runtime.h>
#include <cstdint>
#include <cstddef>

// ---------------- problem constants ----------------
#define BSZ     4
#define LSEQ    2048
#define DIMM    256
#define DINNER  512
#define DSTATE  16
#define DTRANK  16
#define MROWS   (BSZ * LSEQ)          // 8192 rows per direction
#define NXZ     (2 * DINNER)          // 1024
#define NDBC    (DTRANK + 2 * DSTATE) // 48
#define NCHUNK  16                    // parallel-scan chunks over L
#define CHUNK   (LSEQ / NCHUNK)       // 128 steps per chunk
#define NCHAN   (2 * BSZ * DINNER)    // 4096 scan channels (dir,b,d)

// ---------------- input indices (setup_inputs dict order; params in
// _mamba_params insertion order: Win, conv_w, conv_b, Wx, Wdt, bdt, A_log, D, Wout)
#define IDX_X      0
#define IDX_P1     1
#define IDX_P2     10
#define OFF_WIN    0
#define OFF_CONVW  1
#define OFF_CONVB  2
#define OFF_WX     3
#define OFF_WDT    4
#define OFF_BDT    5
#define OFF_ALOG   6
#define OFF_DPAR   7
#define OFF_WOUT   8
#define IDX_LN1W   19
#define IDX_LN1B   20
#define IDX_LN2W   21
#define IDX_LN2B   22
#define IDX_WF     23
#define IDX_BF     24

// ---------------- workspace layout (float offsets) ----------------
#define SZ_XZ_DIR    ((size_t)MROWS * NXZ)     // 8,388,608
#define SZ_XC_DIR    ((size_t)MROWS * DINNER)  // 4,194,304
#define SZ_DBC_DIR   ((size_t)MROWS * NDBC)    //   393,216
#define SZ_DELTA_DIR ((size_t)MROWS * DINNER)
#define SZ_Y_DIR     ((size_t)MROWS * DINNER)
#define SZ_YO_DIR    ((size_t)MROWS * DIMM)    // 2,097,152
#define SZ_XCAT      ((size_t)MROWS * 2 * DIMM)
#define SZ_PQ        ((size_t)NCHAN * NCHUNK * DSTATE)  // 1,048,576

#define WOFF_XZ    ((size_t)0)
#define WOFF_XC    (WOFF_XZ    + 2 * SZ_XZ_DIR)
#define WOFF_DBC   (WOFF_XC    + 2 * SZ_XC_DIR)
#define WOFF_DELTA (WOFF_DBC   + 2 * SZ_DBC_DIR)
#define WOFF_Y     (WOFF_DELTA + 2 * SZ_DELTA_DIR)
#define WOFF_YO    (WOFF_Y     + 2 * SZ_Y_DIR)
#define WOFF_XCAT  (WOFF_YO    + 2 * SZ_YO_DIR)
#define WOFF_P     (WOFF_XCAT  + SZ_XCAT)
#define WOFF_Q     (WOFF_P     + SZ_PQ)
#define WOFF_HI    (WOFF_Q     + SZ_PQ)

typedef float v2f __attribute__((ext_vector_type(2)));
typedef float v8f __attribute__((ext_vector_type(8)));

static __device__ __forceinline__ float silu_f(float x)     { return x / (1.0f + __expf(-x)); }
static __device__ __forceinline__ float softplus_f(float x) { return (x > 20.0f) ? x : log1pf(__expf(x)); }

// K-loop over fp32 WMMA 16x16x4. aRow points at A[src_row(lane)][0] (row-major, lda = K
// contiguous). B row-major KxN. NT adjacent 16-wide N tiles accumulated per wave.
template <int NT>
static __device__ __forceinline__ void wmma_loop_f32(const float* __restrict__ aRow,
                                                     const float* __restrict__ B, int ldb,
                                                     int bcol, int K, int koff, v8f acc[NT]) {
  for (int k0 = 0; k0 < K; k0 += 4) {
    v2f a;
    a.x = aRow[k0 + koff];
    a.y = aRow[k0 + koff + 1];
#pragma unroll
    for (int j = 0; j < NT; ++j) {
      v2f b;
      b.x = B[(size_t)(k0 + koff) * ldb + bcol + 16 * j];
      b.y = B[(size_t)(k0 + koff + 1) * ldb + bcol + 16 * j];
      acc[j] = __builtin_amdgcn_wmma_f32_16x16x4_f32(false, a, false, b, (short)0, acc[j],
                                                     false, false);
    }
  }
}

// ---------------- 1) xz = x(or flip(x)) @ Win ----------------
__global__ __launch_bounds__(32) void k_gemm_in(const float* __restrict__ x,
                                                const float* __restrict__ Win0,
                                                const float* __restrict__ Win1,
                                                float* __restrict__ xz) {
  const int lane = threadIdx.x;
  const int dir  = blockIdx.z;
  const int m0   = blockIdx.y * 16;
  const int n0   = blockIdx.x * 64;

  const int mrow = m0 + (lane & 15);
  const int bb = mrow / LSEQ, tt = mrow % LSEQ;
  const int srow = dir ? (bb * LSEQ + (LSEQ - 1 - tt)) : mrow;
  const float* aRow = x + (size_t)srow * DIMM;
  const float* B = dir ? Win1 : Win0;

  const int koff = (lane >> 4) << 1;
  const int bcol = n0 + (lane & 15);
  v8f acc[4] = {{0,0,0,0,0,0,0,0},{0,0,0,0,0,0,0,0},{0,0,0,0,0,0,0,0},{0,0,0,0,0,0,0,0}};
  wmma_loop_f32<4>(aRow, B, NXZ, bcol, DIMM, koff, acc);

  float* C = xz + (size_t)dir * SZ_XZ_DIR;
  const int crow = m0 + (lane >> 4) * 8;
  const int ccol = n0 + (lane & 15);
#pragma unroll
  for (int j = 0; j < 4; ++j)
#pragma unroll
    for (int i = 0; i < 8; ++i)
      C[(size_t)(crow + i) * NXZ + ccol + 16 * j] = acc[j][i];
}

// ---------------- 2) causal depthwise conv + bias + SiLU ----------------
__global__ __launch_bounds__(256) void k_conv(const float* __restrict__ xz,
                                              const float* __restrict__ cw0,
                                              const float* __restrict__ cw1,
                                              const float* __restrict__ cb0,
                                              const float* __restrict__ cb1,
                                              float* __restrict__ xc) {
  const size_t idx = (size_t)blockIdx.x * 256 + threadIdx.x;
  const int dir = (int)(idx / ((size_t)MROWS * DINNER));
  const size_t rem = idx % ((size_t)MROWS * DINNER);
  const int r = (int)(rem / DINNER);
  const int d = (int)(rem % DINNER);
  const int b = r / LSEQ, t = r % LSEQ;

  const float* cw = (dir ? cw1 : cw0) + d * 4;
  const float* xin = xz + (size_t)dir * SZ_XZ_DIR;  // cols [0,512) of xz
  float acc = (dir ? cb1 : cb0)[d];
#pragma unroll
  for (int j = 0; j < 4; ++j) {
    const int tt = t - 3 + j;
    if (tt >= 0) acc += cw[j] * xin[(size_t)(b * LSEQ + tt) * NXZ + d];
  }
  xc[idx] = silu_f(acc);
}

// ---------------- generic 2-dir WMMA GEMM (plain store) ----------------
template <int NT>
__global__ __launch_bounds__(32) void k_gemm_2dir(const float* __restrict__ A, size_t aDir, int lda,
                                                  const float* __restrict__ B0,
                                                  const float* __restrict__ B1, int ldb,
                                                  float* __restrict__ C, size_t cDir, int ldc,
                                                  int K) {
  const int lane = threadIdx.x;
  const int dir  = blockIdx.z;
  const int m0   = blockIdx.y * 16;
  const int n0   = blockIdx.x * (16 * NT);

  const float* Ad = A + (size_t)dir * aDir;
  const float* Bd = dir ? B1 : B0;
  float* Cd = C + (size_t)dir * cDir;

  const float* aRow = Ad + (size_t)(m0 + (lane & 15)) * lda;
  const int koff = (lane >> 4) << 1;
  const int bcol = n0 + (lane & 15);
  v8f acc[NT];
#pragma unroll
  for (int j = 0; j < NT; ++j) acc[j] = (v8f){0,0,0,0,0,0,0,0};
  wmma_loop_f32<NT>(aRow, Bd, ldb, bcol, K, koff, acc);

  const int crow = m0 + (lane >> 4) * 8;
  const int ccol = n0 + (lane & 15);
#pragma unroll
  for (int j = 0; j < NT; ++j)
#pragma unroll
    for (int i = 0; i < 8; ++i)
      Cd[(size_t)(crow + i) * ldc + ccol + 16 * j] = acc[j][i];
}

// ---------------- 4) delta = softplus(dt_raw @ Wdt + bdt) ----------------
__global__ __launch_bounds__(32) void k_gemm_delta(const float* __restrict__ dbc,
                                                   const float* __restrict__ Wdt0,
                                                   const float* __restrict__ Wdt1,
                                                   const float* __restrict__ bdt0,
                                                   const float* __restrict__ bdt1,
                                                   float* __restrict__ delta) {
  const int lane = threadIdx.x;
  const int dir  = blockIdx.z;
  const int m0   = blockIdx.y * 16;
  const int n0   = blockIdx.x * 64;

  const float* Ad = dbc + (size_t)dir * SZ_DBC_DIR;          // cols [0,16) = dt_raw
  const float* Bd = dir ? Wdt1 : Wdt0;
  const float* bd = dir ? bdt1 : bdt0;
  float* Cd = delta + (size_t)dir * SZ_DELTA_DIR;

  const float* aRow = Ad + (size_t)(m0 + (lane & 15)) * NDBC;
  const int koff = (lane >> 4) << 1;
  const int bcol = n0 + (lane & 15);
  v8f acc[4] = {{0,0,0,0,0,0,0,0},{0,0,0,0,0,0,0,0},{0,0,0,0,0,0,0,0},{0,0,0,0,0,0,0,0}};
  wmma_loop_f32<4>(aRow, Bd, DINNER, bcol, DTRANK, koff, acc);

  const int crow = m0 + (lane >> 4) * 8;
  const int ccol = n0 + (lane & 15);
#pragma unroll
  for (int j = 0; j < 4; ++j)
#pragma unroll
    for (int i = 0; i < 8; ++i) {
      const int col = ccol + 16 * j;
      Cd[(size_t)(crow + i) * DINNER + col] = softplus_f(acc[j][i] + bd[col]);
    }
}

// ---------------- scan helpers ----------------
// channel decomposition: ch in [0,4096): dir = ch>>11, b = (ch>>9)&3, d = ch&511
// chunk summaries P,Q,HI laid out as [(ch*NCHUNK + c)*DSTATE + n]

// 5a) per-chunk composed transfer: h_out = P * h_in + Q
__global__ __launch_bounds__(256) void k_scan_chunks(const float* __restrict__ xc,
                                                     const float* __restrict__ delta,
                                                     const float* __restrict__ dbc,
                                                     const float* __restrict__ alog0,
                                                     const float* __restrict__ alog1,
                                                     float* __restrict__ P,
                                                     float* __restrict__ Q) {
  const int g  = blockIdx.x * 256 + threadIdx.x;  // 65536 = NCHAN * NCHUNK
  const int c  = g >> 12;                          // chunk
  const int ch = g & (NCHAN - 1);                  // channel (lane-consecutive -> coalesced)
  const int dir = ch >> 11;
  const int b   = (ch >> 9) & 3;
  const int d   = ch & (DINNER - 1);

  const float* xc_d = xc + (size_t)dir * SZ_XC_DIR;
  const float* dt_d = delta + (size_t)dir * SZ_DELTA_DIR;
  const float* bc_d = dbc + (size_t)dir * SZ_DBC_DIR;
  const float* alog = (dir ? alog1 : alog0) + d * DSTATE;

  float Ar[DSTATE], Pr[DSTATE], Qr[DSTATE];
#pragma unroll
  for (int n = 0; n < DSTATE; ++n) { Ar[n] = -__expf(alog[n]); Pr[n] = 1.0f; Qr[n] = 0.0f; }

  const int t0 = c * CHUNK;
  for (int t = t0; t < t0 + CHUNK; ++t) {
    const size_t base = (size_t)(b * LSEQ + t);
    const float u  = xc_d[base * DINNER + d];
    const float dt = dt_d[base * DINNER + d];
    const float* bc = bc_d + base * NDBC;
    const float du = dt * u;
#pragma unroll
    for (int n = 0; n < DSTATE; ++n) {
      const float dA = __expf(dt * Ar[n]);
      Pr[n] *= dA;
      Qr[n] = dA * Qr[n] + du * bc[DTRANK + n];
    }
  }
  float* Pp = P + ((size_t)ch * NCHUNK + c) * DSTATE;
  float* Qp = Q + ((size_t)ch * NCHUNK + c) * DSTATE;
#pragma unroll
  for (int n = 0; n < DSTATE; ++n) { Pp[n] = Pr[n]; Qp[n] = Qr[n]; }
}

// 5b) tiny sequential scan over chunk summaries -> initial state per chunk
__global__ __launch_bounds__(256) void k_scan_fix(const float* __restrict__ P,
                                                  const float* __restrict__ Q,
                                                  float* __restrict__ HI) {
  const int ch = blockIdx.x * 256 + threadIdx.x;  // 4096 channels
  float h[DSTATE];
#pragma unroll
  for (int n = 0; n < DSTATE; ++n) h[n] = 0.0f;
  for (int c = 0; c < NCHUNK; ++c) {
    const size_t base = ((size_t)ch * NCHUNK + c) * DSTATE;
#pragma unroll
    for (int n = 0; n < DSTATE; ++n) {
      HI[base + n] = h[n];
      h[n] = P[base + n] * h[n] + Q[base + n];
    }
  }
}

// 5c) re-run each chunk from its true initial state, emit y with fused SiLU gate
__global__ __launch_bounds__(256) void k_scan_emit(const float* __restrict__ xc,
                                                   const float* __restrict__ delta,
                                                   const float* __restrict__ dbc,
                                                   const float* __restrict__ xz,
                                                   const float* __restrict__ alog0,
                                                   const float* __restrict__ alog1,
                                                   const float* __restrict__ Dp0,
                                                   const float* __restrict__ Dp1,
                                                   const float* __restrict__ HI,
                                                   float* __restrict__ y) {
  const int g  = blockIdx.x * 256 + threadIdx.x;  // 65536
  const int c  = g >> 12;
  const int ch = g & (NCHAN - 1);
  const int dir = ch >> 11;
  const int b   = (ch >> 9) & 3;
  const int d   = ch & (DINNER - 1);

  const float* xc_d = xc + (size_t)dir * SZ_XC_DIR;
  const float* dt_d = delta + (size_t)dir * SZ_DELTA_DIR;
  const float* bc_d = dbc + (size_t)dir * SZ_DBC_DIR;
  const float* xz_d = xz + (size_t)dir * SZ_XZ_DIR;
  float* y_d = y + (size_t)dir * SZ_Y_DIR;
  const float* alog = (dir ? alog1 : alog0) + d * DSTATE;
  const float Dd = (dir ? Dp1 : Dp0)[d];

  float Ar[DSTATE], h[DSTATE];
  const float* hi = HI + ((size_t)ch * NCHUNK + c) * DSTATE;
#pragma unroll
  for (int n = 0; n < DSTATE; ++n) { Ar[n] = -__expf(alog[n]); h[n] = hi[n]; }

  const int t0 = c * CHUNK;
  for (int t = t0; t < t0 + CHUNK; ++t) {
    const size_t base = (size_t)(b * LSEQ + t);
    const float u  = xc_d[base * DINNER + d];
    const float dt = dt_d[base * DINNER + d];
    const float* bc = bc_d + base * NDBC;
    const float du = dt * u;
    float yv = 0.0f;
#pragma unroll
    for (int n = 0; n < DSTATE; ++n) {
      const float dA = __expf(dt * Ar[n]);
      h[n] = dA * h[n] + du * bc[DTRANK + n];
      yv += h[n] * bc[DTRANK + DSTATE + n];
    }
    const float z = xz_d[base * NXZ + DINNER + d];
    y_d[base * DINNER + d] = (yv + u * Dd) * silu_f(z);
  }
}

// ---------------- 8) LayerNorm + residual, write concat buffer ----------------
__global__ __launch_bounds__(256) void k_ln_res(const float* __restrict__ yo,
                                                const float* __restrict__ x,
                                                const float* __restrict__ w0,
                                                const float* __restrict__ b0,
                                                const float* __restrict__ w1,
                                                const float* __restrict__ b1,
                                                float* __restrict__ xcat) {
  __shared__ float ssum[8], ssq[8];
  const int r = blockIdx.x;
  const int dir = blockIdx.y;
  const int c = threadIdx.x;

  const float v = yo[(size_t)dir * SZ_YO_DIR + (size_t)r * DIMM + c];
  float s = v, q = v * v;
#pragma unroll
  for (int off = 16; off > 0; off >>= 1) {
    s += __shfl_xor(s, off, 32);
    q += __shfl_xor(q, off, 32);
  }
  if ((c & 31) == 0) { ssum[c >> 5] = s; ssq[c >> 5] = q; }
  __syncthreads();
  if (c == 0) {
    float S = 0.0f, Q = 0.0f;
#pragma unroll
    for (int i = 0; i < 8; ++i) { S += ssum[i]; Q += ssq[i]; }
    const float m = S * (1.0f / DIMM);
    const float var = Q * (1.0f / DIMM) - m * m;
    ssum[0] = m;
    ssq[0] = rsqrtf(var + 1e-5f);
  }
  __syncthreads();
  const float m = ssum[0], rs = ssq[0];
  const float* w = dir ? w1 : w0;
  const float* bb = dir ? b1 : b0;
  const float norm = (v - m) * rs * w[c] + bb[c];

  const int bidx = r >> 11, t = r & (LSEQ - 1);
  const int srct = dir ? (LSEQ - 1 - t) : t;
  const float res = x[(size_t)(bidx * LSEQ + srct) * DIMM + c];
  xcat[(size_t)r * (2 * DIMM) + dir * DIMM + c] = res + norm;
}

// ---------------- 9) out = xcat @ Wf + bf ----------------
__global__ __launch_bounds__(32) void k_gemm_final(const float* __restrict__ xcat,
                                                   const float* __restrict__ Wf,
                                                   const float* __restrict__ bf,
                                                   float* __restrict__ out) {
  const int lane = threadIdx.x;
  const int m0 = blockIdx.y * 16;
  const int n0 = blockIdx.x * 64;

  const float* aRow = xcat + (size_t)(m0 + (lane & 15)) * (2 * DIMM);
  const int koff = (lane >> 4) << 1;
  const int bcol = n0 + (lane & 15);
  v8f acc[4] = {{0,0,0,0,0,0,0,0},{0,0,0,0,0,0,0,0},{0,0,0,0,0,0,0,0},{0,0,0,0,0,0,0,0}};
  wmma_loop_f32<4>(aRow, Wf, DIMM, bcol, 2 * DIMM, koff, acc);

  const int crow = m0 + (lane >> 4) * 8;
  const int ccol = n0 + (lane & 15);
#pragma unroll
  for (int j = 0; j < 4; ++j)
#pragma unroll
    for (int i = 0; i < 8; ++i) {
      const int col = ccol + 16 * j;
      out[(size_t)(crow + i) * DIMM + col] = acc[j][i] + bf[col];
    }
}

// ---------------- host launcher ----------------
extern "C" void kernel_launch(void* const* d_in, const int* in_sizes, int n_in,
                              void* d_out, int out_size, void* d_ws, size_t ws_size,
                              hipStream_t stream) {
  (void)in_sizes; (void)n_in; (void)out_size; (void)ws_size;

  const float* x = (const float*)d_in[IDX_X];
  const float* Win0  = (const float*)d_in[IDX_P1 + OFF_WIN];
  const float* cw0   = (const float*)d_in[IDX_P1 + OFF_CONVW];
  const float* cb0   = (const float*)d_in[IDX_P1 + OFF_CONVB];
  const float* Wx0   = (const float*)d_in[IDX_P1 + OFF_WX];
  const float* Wdt0  = (const float*)d_in[IDX_P1 + OFF_WDT];
  const float* bdt0  = (const float*)d_in[IDX_P1 + OFF_BDT];
  const float* alog0 = (const float*)d_in[IDX_P1 + OFF_ALOG];
  const float* Dp0   = (const float*)d_in[IDX_P1 + OFF_DPAR];
  const float* Wout0 = (const float*)d_in[IDX_P1 + OFF_WOUT];
  const float* Win1  = (const float*)d_in[IDX_P2 + OFF_WIN];
  const float* cw1   = (const float*)d_in[IDX_P2 + OFF_CONVW];
  const float* cb1   = (const float*)d_in[IDX_P2 + OFF_CONVB];
  const float* Wx1   = (const float*)d_in[IDX_P2 + OFF_WX];
  const float* Wdt1  = (const float*)d_in[IDX_P2 + OFF_WDT];
  const float* bdt1  = (const float*)d_in[IDX_P2 + OFF_BDT];
  const float* alog1 = (const float*)d_in[IDX_P2 + OFF_ALOG];
  const float* Dp1   = (const float*)d_in[IDX_P2 + OFF_DPAR];
  const float* Wout1 = (const float*)d_in[IDX_P2 + OFF_WOUT];
  const float* ln1w = (const float*)d_in[IDX_LN1W];
  const float* ln1b = (const float*)d_in[IDX_LN1B];
  const float* ln2w = (const float*)d_in[IDX_LN2W];
  const float* ln2b = (const float*)d_in[IDX_LN2B];
  const float* Wf   = (const float*)d_in[IDX_WF];
  const float* bf   = (const float*)d_in[IDX_BF];

  float* ws = (float*)d_ws;
  float* xz    = ws + WOFF_XZ;
  float* xc    = ws + WOFF_XC;
  float* dbc   = ws + WOFF_DBC;
  float* delta = ws + WOFF_DELTA;
  float* y     = ws + WOFF_Y;
  float* yo    = ws + WOFF_YO;
  float* xcat  = ws + WOFF_XCAT;
  float* Pbuf  = ws + WOFF_P;
  float* Qbuf  = ws + WOFF_Q;
  float* HIbuf = ws + WOFF_HI;

  // 1) xz = x(/flip) @ Win               [8192 x 1024] x 2 dirs
  k_gemm_in<<<dim3(NXZ / 64, MROWS / 16, 2), 32, 0, stream>>>(x, Win0, Win1, xz);
  // 2) depthwise causal conv + SiLU
  k_conv<<<(2 * MROWS * DINNER) / 256, 256, 0, stream>>>(xz, cw0, cw1, cb0, cb1, xc);
  // 3) dbc = xc @ Wx                     [8192 x 48], K=512
  k_gemm_2dir<3><<<dim3(1, MROWS / 16, 2), 32, 0, stream>>>(xc, SZ_XC_DIR, DINNER, Wx0, Wx1,
                                                            NDBC, dbc, SZ_DBC_DIR, NDBC, DINNER);
  // 4) delta = softplus(dt_raw @ Wdt + bdt)   [8192 x 512], K=16
  k_gemm_delta<<<dim3(DINNER / 64, MROWS / 16, 2), 32, 0, stream>>>(dbc, Wdt0, Wdt1, bdt0, bdt1,
                                                                    delta);
  // 5) chunked parallel selective scan (16x parallelism over L) + fused gate
  k_scan_chunks<<<(NCHAN * NCHUNK) / 256, 256, 0, stream>>>(xc, delta, dbc, alog0, alog1,
                                                            Pbuf, Qbuf);
  k_scan_fix<<<NCHAN / 256, 256, 0, stream>>>(Pbuf, Qbuf, HIbuf);
  k_scan_emit<<<(NCHAN * NCHUNK) / 256, 256, 0, stream>>>(xc, delta, dbc, xz, alog0, alog1,
                                                          Dp0, Dp1, HIbuf, y);
  // 7) yo = y @ Wout                     [8192 x 256], K=512
  k_gemm_2dir<4><<<dim3(DIMM / 64, MROWS / 16, 2), 32, 0, stream>>>(y, SZ_Y_DIR, DINNER, Wout0,
                                                                    Wout1, DIMM, yo, SZ_YO_DIR,
                                                                    DIMM, DINNER);
  // 8) LN + residual -> xcat [8192 x 512]
  k_ln_res<<<dim3(MROWS, 2), 256, 0, stream>>>(yo, x, ln1w, ln1b, ln2w, ln2b, xcat);
  // 9) out = xcat @ Wf + bf              [8192 x 256], K=512
  k_gemm_final<<<dim3(DIMM / 64, MROWS / 16, 1), 32, 0, stream>>>(xcat, Wf, bf, (float*)d_out);
}